// GoalKabschVelocityStrategy_43860206026938
// MI455X (gfx1250) — compile-verified
//
#include <hip/hip_runtime.h>
#include <hip/hip_bf16.h>
#include <math.h>

// Problem constants (from reference)
#define B_   2048
#define N_   64
#define D_   512
#define H1_  256
#define G_   8
#define GS_  8
#define E_   128
#define TAUMIN_ 1e-3f
#define EPS_    1e-6f

typedef float v2f __attribute__((ext_vector_type(2)));
typedef float v8f __attribute__((ext_vector_type(8)));
typedef int   v4i_ __attribute__((vector_size(16)));   // matches builtin's pointee type

#define HAS_ASYNC_LDS __has_builtin(__builtin_amdgcn_global_load_async_to_lds_b128)

// ---------------------------------------------------------------------------
// Kernel 1: fused MLP.  h = silu(tokens @ W1 + b1) via WMMA f32 16x16x4,
// then delta = h @ W2 + b2 (tiny GEMM from LDS), x_proto = template + delta.
// One block = 16 rows of (B*N, 512) x (512, 256). 8 waves, each owns two
// 16x16 output tiles (covers all 256 cols). A-panel staged in LDS via the
// CDNA5 async global->LDS path (ASYNCcnt) when available.
// ---------------------------------------------------------------------------
__global__ __launch_bounds__(256)
void mlp_xproto_kernel(const float* __restrict__ tokens,
                       const float* __restrict__ W1,
                       const float* __restrict__ b1,
                       const float* __restrict__ W2,
                       const float* __restrict__ b2,
                       const float* __restrict__ template_xyz,
                       float* __restrict__ x_proto)
{
    __shared__ float a_panel[16 * D_];   // 32 KB
    __shared__ float h_panel[16 * H1_];  // 16 KB

    const int tid  = threadIdx.x;
    const int row0 = blockIdx.x * 16;

#if HAS_ASYNC_LDS
    // Async DMA of the 16x512 fp32 A-panel: 16B per lane per op, straight to
    // LDS, tracked by ASYNCcnt (no VGPR round-trip).
    for (int i = tid * 4; i < 16 * D_; i += 256 * 4) {
        __builtin_amdgcn_global_load_async_to_lds_b128(
            (__attribute__((address_space(1))) v4i_*)(tokens + (size_t)row0 * D_ + i),
            (__attribute__((address_space(3))) v4i_*)(&a_panel[i]),
            /*offset=*/0, /*cpol=*/0);
    }
  #if __has_builtin(__builtin_amdgcn_s_wait_asynccnt)
    __builtin_amdgcn_s_wait_asynccnt(0);
  #else
    asm volatile("s_wait_asynccnt 0" ::: "memory");
  #endif
#else
    {
        const float4* tok4 = (const float4*)(tokens + (size_t)row0 * D_);
        float4*       pan4 = (float4*)a_panel;
        for (int i = tid; i < (16 * D_) / 4; i += 256)
            pan4[i] = tok4[i];
    }
#endif
    __syncthreads();

    const int wave = tid >> 5;
    const int lane = tid & 31;
    const int m    = lane & 15;   // M (and N) position within tile
    const int kh   = lane >> 4;   // which K half-pair this lane holds
    const int n0   = wave * 32;   // this wave covers cols [n0, n0+32)

    v8f c0 = {}; v8f c1 = {};
    for (int kk = 0; kk < D_; kk += 4) {
        const int kl = kk + kh * 2;
        // A fragment (16x4): lanes 0-15 hold K={kk,kk+1}, lanes 16-31 K={kk+2,kk+3}
        v2f a;
        a.x = a_panel[m * D_ + kl];
        a.y = a_panel[m * D_ + kl + 1];
        // B fragments (4x16): same K striping, N across lanes 0-15
        v2f bA, bB;
        bA.x = W1[(size_t)kl       * H1_ + n0 + m];
        bA.y = W1[(size_t)(kl + 1) * H1_ + n0 + m];
        bB.x = W1[(size_t)kl       * H1_ + n0 + 16 + m];
        bB.y = W1[(size_t)(kl + 1) * H1_ + n0 + 16 + m];
        c0 = __builtin_amdgcn_wmma_f32_16x16x4_f32(false, a, false, bA, (short)0, c0, false, false);
        c1 = __builtin_amdgcn_wmma_f32_16x16x4_f32(false, a, false, bB, (short)0, c1, false, false);
    }

    // C/D layout: VGPR r, lanes 0-15 -> M=r, N=lane; lanes 16-31 -> M=r+8
    #pragma unroll
    for (int r = 0; r < 8; ++r) {
        int M = r + kh * 8;
        float x0 = c0[r] + b1[n0 + m];
        float x1 = c1[r] + b1[n0 + 16 + m];
        x0 = x0 / (1.0f + expf(-x0));   // silu
        x1 = x1 / (1.0f + expf(-x1));
        h_panel[M * H1_ + n0 + m]      = x0;
        h_panel[M * H1_ + n0 + 16 + m] = x1;
    }
    __syncthreads();

    // delta = h @ W2 + b2 ; x_proto = template + delta.  48 dot products.
    if (tid < 48) {
        int mrow = tid / 3, j = tid % 3;
        float acc = b2[j];
        for (int k = 0; k < H1_; ++k)
            acc = fmaf(h_panel[mrow * H1_ + k], W2[k * 3 + j], acc);
        int row = row0 + mrow;
        int n   = row & (N_ - 1);
        x_proto[(size_t)row * 3 + j] = template_xyz[n * 3 + j] + acc;
    }
}

// ---------------------------------------------------------------------------
// Kernel 2: per-(b,g) Kabsch via Jacobi eigen of H^T H, det-sign fix,
// then v = (x_goal - keypoints) / tau.  One thread per (b,g).
// group_idx partitions 0..63, so scatter is conflict-free and w == 1.
// ---------------------------------------------------------------------------
__device__ inline void jacobi_rot3(float A[3][3], float V[3][3], int p, int q)
{
    float apq = A[p][q];
    if (fabsf(apq) < 1e-12f) return;
    float th = (A[q][q] - A[p][p]) / (2.f * apq);
    float tt = ((th >= 0.f) ? 1.f : -1.f) / (fabsf(th) + sqrtf(1.f + th * th));
    float c  = 1.f / sqrtf(1.f + tt * tt);
    float s  = tt * c;
    for (int k = 0; k < 3; ++k) {
        float akp = A[k][p], akq = A[k][q];
        A[k][p] = c * akp - s * akq;
        A[k][q] = s * akp + c * akq;
    }
    for (int k = 0; k < 3; ++k) {
        float apk = A[p][k], aqk = A[q][k];
        A[p][k] = c * apk - s * aqk;
        A[q][k] = s * apk + c * aqk;
    }
    for (int k = 0; k < 3; ++k) {
        float vkp = V[k][p], vkq = V[k][q];
        V[k][p] = c * vkp - s * vkq;
        V[k][q] = s * vkp + c * vkq;
    }
}

__global__ __launch_bounds__(256)
void kabsch_velocity_kernel(const float* __restrict__ x_proto,
                            const float* __restrict__ template_xyz,
                            const int*   __restrict__ group_idx,
                            const float* __restrict__ keypoints,
                            const float* __restrict__ timesteps,
                            float* __restrict__ v_buf)
{
    int t = blockIdx.x * blockDim.x + threadIdx.x;
    if (t >= B_ * G_) return;
    int b = t / G_, g = t % G_;

    float src[GS_][3], tgt[GS_][3];
    int   idxs[GS_];
    float mu_s[3] = {0, 0, 0}, mu_t[3] = {0, 0, 0};
    for (int n = 0; n < GS_; ++n) {
        int idx = group_idx[g * GS_ + n];
        idxs[n] = idx;
        #pragma unroll
        for (int i = 0; i < 3; ++i) {
            float s = template_xyz[idx * 3 + i];
            float p = x_proto[((size_t)b * N_ + idx) * 3 + i];
            src[n][i] = s; tgt[n][i] = p;
            mu_s[i] += s; mu_t[i] += p;
        }
    }
    #pragma unroll
    for (int i = 0; i < 3; ++i) { mu_s[i] *= (1.f / GS_); mu_t[i] *= (1.f / GS_); }

    // Cross-covariance H[i][j] = sum_n src_c[n][i] * tgt_c[n][j]
    float Hm[3][3] = {};
    for (int n = 0; n < GS_; ++n)
        #pragma unroll
        for (int i = 0; i < 3; ++i)
            #pragma unroll
            for (int j = 0; j < 3; ++j)
                Hm[i][j] += (src[n][i] - mu_s[i]) * (tgt[n][j] - mu_t[j]);

    // A = H^T H (symmetric)
    float A[3][3];
    #pragma unroll
    for (int i = 0; i < 3; ++i)
        #pragma unroll
        for (int j = 0; j < 3; ++j)
            A[i][j] = Hm[0][i] * Hm[0][j] + Hm[1][i] * Hm[1][j] + Hm[2][i] * Hm[2][j];

    float V[3][3] = {{1, 0, 0}, {0, 1, 0}, {0, 0, 1}};
    for (int sweep = 0; sweep < 8; ++sweep) {
        jacobi_rot3(A, V, 0, 1);
        jacobi_rot3(A, V, 0, 2);
        jacobi_rot3(A, V, 1, 2);
    }
    float w[3] = {A[0][0], A[1][1], A[2][2]};
    // sort eigenpairs descending (columns of V)
    for (int pass = 0; pass < 2; ++pass)
        for (int c = 0; c < 2; ++c)
            if (w[c] < w[c + 1]) {
                float tw = w[c]; w[c] = w[c + 1]; w[c + 1] = tw;
                for (int k = 0; k < 3; ++k) {
                    float tv = V[k][c]; V[k][c] = V[k][c + 1]; V[k][c + 1] = tv;
                }
            }

    // U columns: u_c = H v_c / sigma_c  (fallback: cross product for tiny sigma)
    float U[3][3];
    #pragma unroll
    for (int c = 0; c < 3; ++c) {
        float u[3];
        #pragma unroll
        for (int i = 0; i < 3; ++i)
            u[i] = Hm[i][0] * V[0][c] + Hm[i][1] * V[1][c] + Hm[i][2] * V[2][c];
        float nrm = sqrtf(u[0] * u[0] + u[1] * u[1] + u[2] * u[2]);
        if (nrm > 1e-9f) {
            float inv = 1.f / nrm;
            U[0][c] = u[0] * inv; U[1][c] = u[1] * inv; U[2][c] = u[2] * inv;
        } else {
            // degenerate third direction: right-handed completion
            U[0][c] = U[1][0] * U[2][1] - U[2][0] * U[1][1];
            U[1][c] = U[2][0] * U[0][1] - U[0][0] * U[2][1];
            U[2][c] = U[0][0] * U[1][1] - U[1][0] * U[0][1];
        }
    }

    float detH = Hm[0][0] * (Hm[1][1] * Hm[2][2] - Hm[1][2] * Hm[2][1])
               - Hm[0][1] * (Hm[1][0] * Hm[2][2] - Hm[1][2] * Hm[2][0])
               + Hm[0][2] * (Hm[1][0] * Hm[2][1] - Hm[1][1] * Hm[2][0]);
    float fix = (detH < 0.f) ? -1.f : 1.f;

    // R = v0 u0^T + v1 u1^T + fix * v2 u2^T   (== V diag(1,1,fix) U^T)
    float R[3][3];
    #pragma unroll
    for (int i = 0; i < 3; ++i)
        #pragma unroll
        for (int j = 0; j < 3; ++j)
            R[i][j] = V[i][0] * U[j][0] + V[i][1] * U[j][1] + fix * V[i][2] * U[j][2];

    float tv[3];
    #pragma unroll
    for (int i = 0; i < 3; ++i)
        tv[i] = mu_t[i] - (R[i][0] * mu_s[0] + R[i][1] * mu_s[1] + R[i][2] * mu_s[2]);

    float tau     = fmaxf(1.f - timesteps[b], TAUMIN_);
    float inv_tau = 1.f / tau;

    for (int n = 0; n < GS_; ++n) {
        int idx = idxs[n];
        #pragma unroll
        for (int i = 0; i < 3; ++i) {
            float proj = R[i][0] * src[n][0] + R[i][1] * src[n][1]
                       + R[i][2] * src[n][2] + tv[i];
            size_t o = ((size_t)b * N_ + idx) * 3 + i;
            v_buf[o] = (proj - keypoints[o]) * inv_tau;
        }
    }
}

// ---------------------------------------------------------------------------
// Kernel 3: tangent projection.  One block per batch item.
// Builds A = 4*edge_dot*sign + eps*I (128x128) in dynamic LDS (~70 KB,
// fine on CDNA5's 320 KB/WGP), solves A lam = rhs by Gaussian elimination,
// scatters constraint forces with LDS float atomics.
// ---------------------------------------------------------------------------
__global__ __launch_bounds__(256)
void tangent_kernel(const float* __restrict__ keypoints,
                    const float* __restrict__ v_buf,
                    const int*   __restrict__ edge_index,
                    float* __restrict__ out)
{
    extern __shared__ float smem[];
    const int LDA = E_ + 1;                 // 129 to de-conflict banks
    float* As    = smem;                    // E_*LDA
    float* dxs   = As + E_ * LDA;           // E_*3
    float* rhs   = dxs + E_ * 3;            // E_
    float* lam   = rhs + E_;                // E_
    float* fv    = lam + E_;                // E_  (elimination factors)
    float* vcorr = fv + E_;                 // N_*3

    __shared__ int s_src[E_], s_dst[E_];

    const int tid = threadIdx.x;
    const int b   = blockIdx.x;
    const float* kp = keypoints + (size_t)b * N_ * 3;
    const float* v  = v_buf     + (size_t)b * N_ * 3;

    if (tid < E_) {
        int s = edge_index[tid];
        int d = edge_index[E_ + tid];
        s_src[tid] = s; s_dst[tid] = d;
        float dx0 = kp[s * 3 + 0] - kp[d * 3 + 0];
        float dx1 = kp[s * 3 + 1] - kp[d * 3 + 1];
        float dx2 = kp[s * 3 + 2] - kp[d * 3 + 2];
        dxs[tid * 3 + 0] = dx0; dxs[tid * 3 + 1] = dx1; dxs[tid * 3 + 2] = dx2;
        float dv0 = v[s * 3 + 0] - v[d * 3 + 0];
        float dv1 = v[s * 3 + 1] - v[d * 3 + 1];
        float dv2 = v[s * 3 + 2] - v[d * 3 + 2];
        rhs[tid] = 2.f * (dx0 * dv0 + dx1 * dv1 + dx2 * dv2);
    }
    if (tid < N_ * 3) vcorr[tid] = 0.f;
    __syncthreads();

    for (int q = tid; q < E_ * E_; q += 256) {
        int e = q >> 7, f = q & (E_ - 1);
        float dot = dxs[e * 3] * dxs[f * 3] + dxs[e * 3 + 1] * dxs[f * 3 + 1]
                  + dxs[e * 3 + 2] * dxs[f * 3 + 2];
        int se = s_src[e], de = s_dst[e], sf = s_src[f], df = s_dst[f];
        float sign = (float)((se == sf) - (se == df) - (de == sf) + (de == df));
        float a = 4.f * dot * sign;
        if (e == f) a += EPS_;
        As[e * LDA + f] = a;
    }
    __syncthreads();

    // Forward elimination (no pivoting; diag = 8|dx|^2 + eps dominates)
    for (int k = 0; k < E_ - 1; ++k) {
        int rem = E_ - 1 - k;               // rows below pivot
        if (tid < rem) {
            int i = k + 1 + tid;
            fv[i] = As[i * LDA + k] / As[k * LDA + k];
        }
        __syncthreads();
        int cnt  = rem + 1;                 // cols k+1..127 plus rhs slot
        int flat = rem * cnt;
        for (int q = tid; q < flat; q += 256) {
            int i  = k + 1 + q / cnt;
            int jj = q % cnt;
            float f = fv[i];
            if (jj < rem) {
                int j = k + 1 + jj;
                As[i * LDA + j] -= f * As[k * LDA + j];
            } else {
                rhs[i] -= f * rhs[k];
            }
        }
        __syncthreads();
    }

    // Back substitution
    for (int k = E_ - 1; k >= 0; --k) {
        if (tid == 0) lam[k] = rhs[k] / As[k * LDA + k];
        __syncthreads();
        if (tid < k) rhs[tid] -= As[tid * LDA + k] * lam[k];
        __syncthreads();
    }

    // force = 2*lam*dx ; scatter +force to src node, -force to dst node
    if (tid < E_) {
        float l2 = 2.f * lam[tid];
        float fx = l2 * dxs[tid * 3 + 0];
        float fy = l2 * dxs[tid * 3 + 1];
        float fz = l2 * dxs[tid * 3 + 2];
        int s = s_src[tid], d = s_dst[tid];
        atomicAdd(&vcorr[s * 3 + 0],  fx);
        atomicAdd(&vcorr[s * 3 + 1],  fy);
        atomicAdd(&vcorr[s * 3 + 2],  fz);
        atomicAdd(&vcorr[d * 3 + 0], -fx);
        atomicAdd(&vcorr[d * 3 + 1], -fy);
        atomicAdd(&vcorr[d * 3 + 2], -fz);
    }
    __syncthreads();

    if (tid < N_ * 3)
        out[(size_t)b * N_ * 3 + tid] = v[tid] - vcorr[tid];
}

// ---------------------------------------------------------------------------
extern "C" void kernel_launch(void* const* d_in, const int* in_sizes, int n_in,
                              void* d_out, int out_size, void* d_ws, size_t ws_size,
                              hipStream_t stream)
{
    const float* keypoints    = (const float*)d_in[0];
    const float* timesteps    = (const float*)d_in[1];
    const float* tokens       = (const float*)d_in[2];
    const float* template_xyz = (const float*)d_in[3];
    const float* W1           = (const float*)d_in[4];
    const float* b1           = (const float*)d_in[5];
    const float* W2           = (const float*)d_in[6];
    const float* b2           = (const float*)d_in[7];
    const int*   group_idx    = (const int*)d_in[8];
    const int*   edge_index   = (const int*)d_in[9];
    float*       out          = (float*)d_out;

    float* x_proto = (float*)d_ws;                       // B*N*3 floats
    float* v_buf   = x_proto + (size_t)B_ * N_ * 3;      // B*N*3 floats

    mlp_xproto_kernel<<<(B_ * N_) / 16, 256, 0, stream>>>(
        tokens, W1, b1, W2, b2, template_xyz, x_proto);

    kabsch_velocity_kernel<<<(B_ * G_ + 255) / 256, 256, 0, stream>>>(
        x_proto, template_xyz, group_idx, keypoints, timesteps, v_buf);

    size_t smem_bytes = (size_t)(E_ * (E_ + 1) + E_ * 3 + E_ + E_ + E_ + N_ * 3)
                        * sizeof(float);
    tangent_kernel<<<B_, 256, smem_bytes, stream>>>(keypoints, v_buf, edge_index, out);
}